// BatchedNMS_77154792505966
// MI455X (gfx1250) — compile-verified
//
#include <hip/hip_runtime.h>
#include <hip/hip_bf16.h>
#include <stdint.h>

#define N_BOX 8192
#define MASK_WORDS (N_BOX / 32) /* 256 */

typedef float v2f __attribute__((ext_vector_type(2)));
typedef float v8f __attribute__((ext_vector_type(8)));

#if defined(__has_builtin)
#if __has_builtin(__builtin_amdgcn_wmma_f32_16x16x4_f32)
#define HAVE_WMMA_F32X4 1
#endif
#endif

// ---------------------------------------------------------------- kernel 1
// boxes_for_nms[i] = bbox[i] + class[i] * (MAX_COORD + 1)
__global__ void prep_boxes(const float* __restrict__ bboxes,
                           const int* __restrict__ cls,
                           float4* __restrict__ boxes_nms) {
    int i = blockIdx.x * blockDim.x + threadIdx.x;
    if (i < N_BOX) {
        const float4* bb4 = (const float4*)bboxes;
        float4 b = bb4[i];
        float off = (float)cls[i] * 2049.0f;
        b.x += off; b.y += off; b.z += off; b.w += off;
        boxes_nms[i] = b;
    }
}

// ---------------------------------------------------------------- kernel 2
// Stable argsort(-scores) via bitonic sort of u64 keys in LDS (64KB of the
// 320KB WGP LDS). Key = score_bits<<32 | (N-1-i): descending sort gives
// descending score, ties broken by ascending original index (stable).
__global__ __launch_bounds__(1024) void sort_gather(
        const float* __restrict__ scores,
        const float4* __restrict__ boxes_nms,
        int* __restrict__ order,
        float4* __restrict__ sboxes) {
    __shared__ unsigned long long key[N_BOX];
    int t = threadIdx.x;
    for (int e = 0; e < N_BOX / 1024; ++e) {
        int i = t + e * 1024;
        unsigned sb = __float_as_uint(scores[i]); // scores in [0,1): monotonic
        key[i] = ((unsigned long long)sb << 32) |
                 (unsigned long long)(unsigned)(N_BOX - 1 - i);
    }
    __syncthreads();
    for (int k = 2; k <= N_BOX; k <<= 1) {
        for (int j = k >> 1; j > 0; j >>= 1) {
            for (int e = 0; e < N_BOX / 1024; ++e) {
                int i = t + e * 1024;
                int ixj = i ^ j;
                if (ixj > i) {
                    unsigned long long a = key[i], b = key[ixj];
                    bool desc = ((i & k) == 0);
                    if (desc ? (a < b) : (a > b)) { key[i] = b; key[ixj] = a; }
                }
            }
            __syncthreads();
        }
    }
    for (int e = 0; e < N_BOX / 1024; ++e) {
        int i = t + e * 1024;
        unsigned long long kk = key[i];
        int src = (N_BOX - 1) - (int)(unsigned)(kk & 0xFFFFFFFFull);
        order[i] = src;
        sboxes[i] = boxes_nms[src];
    }
}

// ---------------------------------------------------------------- kernel 3
// Suppression bitmask. One wave32 per 16-row x 32-col tile.
// WMMA f32 16x16x4 computes D[m][n] = area_row[m] + area_col[n]
// (A = [area_r | 1 | 0 | 0], B = [1 ; area_c ; 0 ; 0]).
// The WMMA C/D layout (lane L, vgpr r -> row r+8*(L>=16), col L&15) lines up
// exactly with ballot_w32: one compare per accumulator VGPR yields two 16-bit
// row half-masks. IoU test is division-free: union > 0 always, so
// inter/union > thr  <=>  inter > thr * union.
__global__ __launch_bounds__(256) void mask_kernel(
        const float4* __restrict__ sb,
        const float* __restrict__ thrp,
        unsigned* __restrict__ mask) {
    const float thr = *thrp;
    const int wave = threadIdx.x >> 5;
    const int lane = threadIdx.x & 31;
    const int tile = blockIdx.x * 8 + wave;
    const int rbase = (tile >> 8) << 4;  // 512 row blocks of 16
    const int cbase = (tile & 255) << 5; // 256 col blocks of 32
    const int cw = cbase >> 5;

    // Only forward bits (col > row) can be set. If impossible, store zeros.
    if (cbase + 31 <= rbase) {
        if (lane < 16) mask[(rbase + lane) * MASK_WORDS + cw] = 0u;
        return;
    }

    __builtin_prefetch(sb + cbase, 0, 0); // global_prefetch_b8

    const bool lo16 = lane < 16;
    float4 rb  = sb[rbase + (lane & 15)];       // row boxes, lanes 0-15 valid
    float4 cbl = sb[cbase + (lane & 15)];       // col boxes 0..15
    float4 cbh = sb[cbase + 16 + (lane & 15)];  // col boxes 16..31
    float ar = (rb.z - rb.x) * (rb.w - rb.y);
    float al = (cbl.z - cbl.x) * (cbl.w - cbl.y);
    float ah = (cbh.z - cbh.x) * (cbh.w - cbh.y);

#if defined(HAVE_WMMA_F32X4)
    v2f A;  A[0]  = lo16 ? ar   : 0.0f; A[1]  = lo16 ? 1.0f : 0.0f;
    v2f Bl; Bl[0] = lo16 ? 1.0f : 0.0f; Bl[1] = lo16 ? al   : 0.0f;
    v2f Bh; Bh[0] = lo16 ? 1.0f : 0.0f; Bh[1] = lo16 ? ah   : 0.0f;
    v8f z = {};
    v8f dlo = __builtin_amdgcn_wmma_f32_16x16x4_f32(
        false, A, false, Bl, (short)0, z, false, false);
    v8f dhi = __builtin_amdgcn_wmma_f32_16x16x4_f32(
        false, A, false, Bh, (short)0, z, false, false);
#endif

    const int cl = cbase + (lane & 15);
    const int ch = cl + 16;
#pragma unroll
    for (int r = 0; r < 8; ++r) {
        int m = r + (lo16 ? 0 : 8); // this lane's row within the tile
        float rx1 = __shfl(rb.x, m, 32);
        float ry1 = __shfl(rb.y, m, 32);
        float rx2 = __shfl(rb.z, m, 32);
        float ry2 = __shfl(rb.w, m, 32);
        int rowIdx = rbase + m;
#if defined(HAVE_WMMA_F32X4)
        float slo = dlo[r];
        float shi = dhi[r];
#else
        float arm = __shfl(ar, m, 32);
        float slo = arm + al;
        float shi = arm + ah;
#endif
        float iwl = fmaxf(fminf(rx2, cbl.z) - fmaxf(rx1, cbl.x), 0.0f);
        float ihl = fmaxf(fminf(ry2, cbl.w) - fmaxf(ry1, cbl.y), 0.0f);
        float il  = iwl * ihl;
        // inter/(sum-inter) > thr  <=>  inter > thr*(sum-inter)   (union > 0)
        bool  pl  = (il > thr * (slo - il)) && (cl > rowIdx);

        float iwh = fmaxf(fminf(rx2, cbh.z) - fmaxf(rx1, cbh.x), 0.0f);
        float ihh = fmaxf(fminf(ry2, cbh.w) - fmaxf(ry1, cbh.y), 0.0f);
        float ih  = iwh * ihh;
        bool  ph  = (ih > thr * (shi - ih)) && (ch > rowIdx);

        unsigned blo = __builtin_amdgcn_ballot_w32(pl);
        unsigned bhi = __builtin_amdgcn_ballot_w32(ph);
        if (lane == 0) {
            // bits[15:0] of blo/bhi: row r ; bits[31:16]: row r+8
            unsigned wl = (blo & 0xFFFFu) | ((bhi & 0xFFFFu) << 16);
            unsigned wh = (blo >> 16) | (bhi & 0xFFFF0000u);
            mask[(rbase + r) * MASK_WORDS + cw] = wl;
            mask[(rbase + r + 8) * MASK_WORDS + cw] = wh;
        }
    }
}

// ---------------------------------------------------------------- kernel 4
// Serial greedy scan; mask has only forward bits so final keep = ~removed.
__global__ __launch_bounds__(256) void greedy_scan(
        const unsigned* __restrict__ mask,
        unsigned* __restrict__ keepw) {
    __shared__ unsigned rem[MASK_WORDS];
    int t = threadIdx.x;
    rem[t] = 0u;
    __syncthreads();
    for (int i = 0; i < N_BOX; ++i) {
        unsigned w = rem[i >> 5];
        __syncthreads(); // all reads before any writes
        if (!((w >> (i & 31)) & 1u)) {
            rem[t] |= mask[i * MASK_WORDS + t];
        }
        __syncthreads();
    }
    keepw[t] = ~rem[t];
}

// ---------------------------------------------------------------- kernel 5
__global__ void zero_out(float* __restrict__ out, int n) {
    int i = blockIdx.x * blockDim.x + threadIdx.x;
    if (i < n) out[i] = 0.0f;
}

// ---------------------------------------------------------------- kernel 6
// Prefix-scan of keep bits, scatter kept detections (score order) to front.
__global__ __launch_bounds__(1024) void compact(
        const unsigned* __restrict__ keepw,
        const int* __restrict__ order,
        const float* __restrict__ det,
        float* __restrict__ out) {
    __shared__ unsigned cnt[1024];
    int t = threadIdx.x;
    int p0 = t * 8; // 8 contiguous bits, never straddles a mask word
    unsigned bits = (keepw[p0 >> 5] >> (p0 & 31)) & 0xFFu;
    cnt[t] = (unsigned)__popc(bits);
    __syncthreads();
    for (int off = 1; off < 1024; off <<= 1) {
        unsigned v = cnt[t];
        unsigned add = (t >= off) ? cnt[t - off] : 0u;
        __syncthreads();
        cnt[t] = v + add;
        __syncthreads();
    }
    unsigned base = cnt[t] - (unsigned)__popc(bits); // exclusive prefix
    for (int e = 0; e < 8; ++e) {
        if ((bits >> e) & 1u) {
            int p = p0 + e;
            int src = order[p];
            int dst = (int)(base++);
#pragma unroll
            for (int c = 0; c < 6; ++c) out[dst * 6 + c] = det[src * 6 + c];
        }
    }
}

// ----------------------------------------------------------------
extern "C" void kernel_launch(void* const* d_in, const int* in_sizes, int n_in,
                              void* d_out, int out_size, void* d_ws, size_t ws_size,
                              hipStream_t stream) {
    const float* det    = (const float*)d_in[0]; // (1, N, 6) f32
    const int*   cls    = (const int*)d_in[1];   // (N,) i32
    const float* bboxes = (const float*)d_in[2]; // (N, 4) f32
    const float* scores = (const float*)d_in[3]; // (N,) f32
    const float* thr    = (const float*)d_in[4]; // scalar f32
    float* out = (float*)d_out;

    char* ws = (char*)d_ws;
    float4*   boxes_nms = (float4*)(ws);                       // 128 KB
    float4*   sboxes    = (float4*)(ws + 131072);              // 128 KB
    int*      order     = (int*)(ws + 262144);                 // 32 KB
    unsigned* keepw     = (unsigned*)(ws + 262144 + 32768);    // 1 KB
    unsigned* mask      = (unsigned*)(ws + 299008);            // 8 MB

    prep_boxes<<<N_BOX / 256, 256, 0, stream>>>(bboxes, cls, boxes_nms);
    sort_gather<<<1, 1024, 0, stream>>>(scores, boxes_nms, order, sboxes);
    // (8192/16 row tiles) * (8192/32 col tiles) / 8 waves per block
    mask_kernel<<<(N_BOX / 16) * (N_BOX / 32) / 8, 256, 0, stream>>>(sboxes, thr, mask);
    greedy_scan<<<1, 256, 0, stream>>>(mask, keepw);
    zero_out<<<(N_BOX * 6 + 255) / 256, 256, 0, stream>>>(out, N_BOX * 6);
    compact<<<1, 1024, 0, stream>>>(keepw, order, det, out);
}